// KinematicL2_35613868818702
// MI455X (gfx1250) — compile-verified
//
#include <hip/hip_runtime.h>
#include <hip/hip_bf16.h>
#include <stdint.h>

// MI455X plan:
//  - 48MB streamed @23.3TB/s => ~2.1us memory floor; per-pair SE(3) math is
//    fp32 VALU (no 16x16 matrix structure), so stage1 = async-LDS staging +
//    register math; stage2 = WMMA-chained deterministic final reduction.
//  - score^2 == dot(xi,xi): the norm's sqrt cancels, skip it.

#define DT_HALF 0.05f  // DT/2 with DT = 0.1

typedef float v2f __attribute__((ext_vector_type(2)));
typedef float v8f __attribute__((ext_vector_type(8)));
typedef int   vs4i __attribute__((vector_size(16)));  // matches builtin param

// ---------------- async global->LDS helpers (CDNA5 ASYNCcnt path) ----------
#if defined(__gfx1250__)
#  if __has_builtin(__builtin_amdgcn_global_load_async_to_lds_b128)
#    define ASYNC_COPY_B128(gp, lp)                                           \
       __builtin_amdgcn_global_load_async_to_lds_b128(                        \
           (__attribute__((address_space(1))) vs4i*)(uintptr_t)(gp),          \
           (__attribute__((address_space(3))) vs4i*)(uintptr_t)(lp), 0, 0)
#  else
#    define ASYNC_COPY_B128(gp, lp)                                           \
       asm volatile("global_load_async_to_lds_b128 %0, %1, off"               \
                    :: "v"((unsigned)(uintptr_t)(lp)), "v"(gp) : "memory")
#  endif
#  if __has_builtin(__builtin_amdgcn_s_wait_asynccnt)
#    define WAIT_ASYNC() __builtin_amdgcn_s_wait_asynccnt(0)
#  else
#    define WAIT_ASYNC() asm volatile("s_wait_asynccnt 0" ::: "memory")
#  endif
#  define USE_ASYNC_LDS 1
#else
#  define USE_ASYNC_LDS 0
#endif

// ---------------- small SE(3) helpers (row-major 3x3) ----------------------
__device__ __forceinline__ void se3_exp6(const float* __restrict__ tau,
                                         const float* __restrict__ phi,
                                         float R[9], float t[3]) {
  const float x = phi[0], y = phi[1], z = phi[2];
  const float th2 = x * x + y * y + z * z;
  const float th  = sqrtf(th2);
  float A, B, C;
  if (th < 1e-4f) {
    A = 1.f - th2 * (1.f / 6.f);
    B = 0.5f - th2 * (1.f / 24.f);
    C = (1.f / 6.f) - th2 * (1.f / 120.f);
  } else {
    const float s = __sinf(th), c = __cosf(th);
    const float inv = 1.f / th;
    A = s * inv;
    B = (1.f - c) * inv * inv;
    C = (th - s) * inv * inv * inv;
  }
  const float xx = x * x, yy = y * y, zz = z * z;
  const float xy = x * y, xz = x * z, yz = y * z;
  // K2 = phi*phi^T - th2*I ;  R = I + A*K + B*K2 ; V = I + B*K + C*K2
  R[0] = 1.f + B * (xx - th2); R[1] = B * xy - A * z;      R[2] = B * xz + A * y;
  R[3] = B * xy + A * z;       R[4] = 1.f + B * (yy - th2); R[5] = B * yz - A * x;
  R[6] = B * xz - A * y;       R[7] = B * yz + A * x;      R[8] = 1.f + B * (zz - th2);
  const float V0 = 1.f + C * (xx - th2), V1 = C * xy - B * z, V2 = C * xz + B * y;
  const float V3 = C * xy + B * z, V4 = 1.f + C * (yy - th2), V5 = C * yz - B * x;
  const float V6 = C * xz - B * y, V7 = C * yz + B * x, V8 = 1.f + C * (zz - th2);
  t[0] = V0 * tau[0] + V1 * tau[1] + V2 * tau[2];
  t[1] = V3 * tau[0] + V4 * tau[1] + V5 * tau[2];
  t[2] = V6 * tau[0] + V7 * tau[1] + V8 * tau[2];
}

__device__ __forceinline__ void mm3(const float* A, const float* B, float* C) {
#pragma unroll
  for (int i = 0; i < 3; ++i)
#pragma unroll
    for (int j = 0; j < 3; ++j)
      C[i * 3 + j] = A[i * 3 + 0] * B[0 + j] + A[i * 3 + 1] * B[3 + j] +
                     A[i * 3 + 2] * B[6 + j];
}

__device__ __forceinline__ void mmT3(const float* A, const float* B, float* C) {
  // C = A * B^T
#pragma unroll
  for (int i = 0; i < 3; ++i)
#pragma unroll
    for (int j = 0; j < 3; ++j)
      C[i * 3 + j] = A[i * 3 + 0] * B[j * 3 + 0] + A[i * 3 + 1] * B[j * 3 + 1] +
                     A[i * 3 + 2] * B[j * 3 + 2];
}

__device__ __forceinline__ void mv3(const float* A, const float* v, float* o) {
  o[0] = A[0] * v[0] + A[1] * v[1] + A[2] * v[2];
  o[1] = A[3] * v[0] + A[4] * v[1] + A[5] * v[2];
  o[2] = A[6] * v[0] + A[7] * v[1] + A[8] * v[2];
}

// loss = |se3_log(fwd * inv(bwd))|^2 for one adjacent pair
__device__ __forceinline__ float pair_loss(const float* __restrict__ x0,
                                           const float* __restrict__ x1) {
  float Rh[9], th[3], Re[9], te[3], Rf[9], tf[3], Rb[9], tb[3], w[6];

  // fwd = exp(x0[6:12]*dt/2) o exp(x0[0:6])
  se3_exp6(x0, x0 + 3, Rh, th);
#pragma unroll
  for (int j = 0; j < 6; ++j) w[j] = x0[6 + j] * DT_HALF;
  se3_exp6(w, w + 3, Re, te);
  mm3(Re, Rh, Rf);
  mv3(Re, th, tf);
  tf[0] += te[0]; tf[1] += te[1]; tf[2] += te[2];

  // bwd = exp(x1[6:12]*(-dt/2)) o exp(x1[0:6])
  se3_exp6(x1, x1 + 3, Rh, th);
#pragma unroll
  for (int j = 0; j < 6; ++j) w[j] = x1[6 + j] * (-DT_HALF);
  se3_exp6(w, w + 3, Re, te);
  mm3(Re, Rh, Rb);
  mv3(Re, th, tb);
  tb[0] += te[0]; tb[1] += te[1]; tb[2] += te[2];

  // rel = fwd o inv(bwd): Rd = Rf*Rb^T ; td = tf - Rd*tb
  float Rd[9], rt[3], td[3];
  mmT3(Rf, Rb, Rd);
  mv3(Rd, tb, rt);
  td[0] = tf[0] - rt[0]; td[1] = tf[1] - rt[1]; td[2] = tf[2] - rt[2];

  // so3_log
  const float tr = Rd[0] + Rd[4] + Rd[8];
  const float cth = fminf(fmaxf((tr - 1.f) * 0.5f, -1.f + 1e-7f), 1.f - 1e-7f);
  const float ang = acosf(cth);
  const float ang2 = ang * ang;
  float coef, D;
  if (ang < 1e-4f) {
    coef = 0.5f + ang2 * (1.f / 12.f);
    D = (1.f / 12.f) + ang2 * (1.f / 720.f);
  } else {
    const float s = __sinf(ang), c = __cosf(ang);
    coef = ang / (2.f * s);
    D = (1.f - ang * s / (2.f * (1.f - c))) / ang2;
  }
  const float px = coef * (Rd[7] - Rd[5]);
  const float py = coef * (Rd[2] - Rd[6]);
  const float pz = coef * (Rd[3] - Rd[1]);
  const float pp = px * px + py * py + pz * pz;
  // Vinv = I - 0.5K + D*(p p^T - pp I)
  const float Vi0 = 1.f + D * (px * px - pp), Vi1 = 0.5f * pz + D * px * py,
              Vi2 = -0.5f * py + D * px * pz;
  const float Vi3 = -0.5f * pz + D * px * py, Vi4 = 1.f + D * (py * py - pp),
              Vi5 = 0.5f * px + D * py * pz;
  const float Vi6 = 0.5f * py + D * px * pz, Vi7 = -0.5f * px + D * py * pz,
              Vi8 = 1.f + D * (pz * pz - pp);
  const float u0 = Vi0 * td[0] + Vi1 * td[1] + Vi2 * td[2];
  const float u1 = Vi3 * td[0] + Vi4 * td[1] + Vi5 * td[2];
  const float u2 = Vi6 * td[0] + Vi7 * td[1] + Vi8 * td[2];
  return u0 * u0 + u1 * u1 + u2 * u2 + pp;  // |xi|^2 (no sqrt needed)
}

// --------------- stage 1: one block per trajectory -------------------------
__global__ __launch_bounds__(128) void k_stage1(
    const float* __restrict__ traj, const int* __restrict__ kidx,
    const float* __restrict__ tw, const float* __restrict__ kw,
    float* __restrict__ partials, int T) {
  __shared__ float sx[128 * 12];   // one trajectory, 6 KB
  __shared__ float sred[128];
  const int b = blockIdx.x;
  const int tid = threadIdx.x;

  const float4* g4 = (const float4*)(traj + (size_t)b * (size_t)T * 12u);
  float4* s4 = (float4*)sx;
  const int nvec = (T * 12) / 4;  // 384 for T=128

#if USE_ASYNC_LDS
  for (int i = tid; i < nvec; i += 128) {
    ASYNC_COPY_B128(g4 + i, s4 + i);
  }
  WAIT_ASYNC();
  __syncthreads();
#else
  for (int i = tid; i < nvec; i += 128) s4[i] = g4[i];
  __syncthreads();
#endif

  float c = 0.f;
  if (tid < T - 1)
    c = pair_loss(sx + tid * 12, sx + (tid + 1) * 12) * tw[tid];

  sred[tid] = c;
  __syncthreads();
#pragma unroll
  for (int off = 64; off >= 1; off >>= 1) {
    if (tid < off) sred[tid] += sred[tid + off];
    __syncthreads();
  }
  if (tid == 0)
    partials[b] = sred[0] * (1.f / (float)(T - 1)) * kw[kidx[b]];
}

// --------------- stage 2: WMMA-chained deterministic final sum -------------
__global__ __launch_bounds__(32) void k_stage2(const float* __restrict__ p,
                                               float* __restrict__ out, int n) {
  const int lane = threadIdx.x;
#if defined(__gfx1250__) && __has_builtin(__builtin_amdgcn_wmma_f32_16x16x4_f32)
  // A = 16x4 chunk of partials, B = ones(4x16) => D rows accumulate row-sums.
  // EXEC stays all-1s through the whole WMMA loop (no per-lane guards):
  // the multiple-of-64 body does straight b64 loads; the tail (empty for
  // n=8192) is summed lane-parallel afterwards.
  v8f acc = {0.f, 0.f, 0.f, 0.f, 0.f, 0.f, 0.f, 0.f};
  const v2f bones = {1.f, 1.f};
  const int nfull = n & ~63;
  // A[m,k]: m = lane&15, k = 2*(lane>>4) + {0,1} -> contiguous float pair
  const int off0 = (lane & 15) * 4 + (lane >> 4) * 2;
  for (int base = 0; base < nfull; base += 64) {
    const v2f a = *(const v2f*)(p + base + off0);  // global_load_b64
    acc = __builtin_amdgcn_wmma_f32_16x16x4_f32(false, a, false, bones,
                                                (short)0, acc, false, false);
  }
  // VGPR r holds M = r + 8*(lane>=16); every column identical -> lo+hi halves
  float s = acc[0] + acc[1] + acc[2] + acc[3] + acc[4] + acc[5] + acc[6] + acc[7];
  s += __shfl_xor(s, 16, 32);
  // lane-parallel tail (n % 64 leftovers), then full wave reduction
  float tail = 0.f;
  for (int i = nfull + lane; i < n; i += 32) tail += p[i];
#pragma unroll
  for (int m = 16; m >= 1; m >>= 1) tail += __shfl_xor(tail, m, 32);
  if (lane == 0) out[0] = (s + tail) / (float)n;
#else
  if (lane == 0) {
    float s = 0.f;
    for (int i = 0; i < n; ++i) s += p[i];
    out[0] = s / (float)n;
  }
#endif
}

extern "C" void kernel_launch(void* const* d_in, const int* in_sizes, int n_in,
                              void* d_out, int out_size, void* d_ws, size_t ws_size,
                              hipStream_t stream) {
  const float* traj = (const float*)d_in[0];   // (8192, 128, 12) f32
  const int*   kidx = (const int*)d_in[1];     // (8192,) i32
  const float* tw   = (const float*)d_in[2];   // (127,) f32
  const float* kw   = (const float*)d_in[3];   // (1000,) f32
  const int n_traj = in_sizes[1];              // 8192
  const int T      = in_sizes[2] + 1;          // 127 + 1 = 128

  float* partials = (float*)d_ws;              // n_traj floats (32 KB)
  k_stage1<<<dim3(n_traj), dim3(128), 0, stream>>>(traj, kidx, tw, kw,
                                                   partials, T);
  k_stage2<<<dim3(1), dim3(32), 0, stream>>>(partials, (float*)d_out, n_traj);
}